// Attention_31585189494995
// MI455X (gfx1250) — compile-verified
//
#include <hip/hip_runtime.h>

#define B_    32
#define H_    32
#define HKV_  8
#define G_    4
#define DH_   128
#define MAXL_ 2048
#define NW    8      // waves per block
#define PS    20     // padded LDS stride for P tile (floats) -> conflict-free

typedef __attribute__((ext_vector_type(2))) float v2f;
typedef __attribute__((ext_vector_type(8))) float v8f;

__device__ __forceinline__ float hmax16(float x) {
  x = fmaxf(x, __shfl_xor(x, 1));
  x = fmaxf(x, __shfl_xor(x, 2));
  x = fmaxf(x, __shfl_xor(x, 4));
  x = fmaxf(x, __shfl_xor(x, 8));
  return x;
}
__device__ __forceinline__ float hsum16(float x) {
  x += __shfl_xor(x, 1);
  x += __shfl_xor(x, 2);
  x += __shfl_xor(x, 4);
  x += __shfl_xor(x, 8);
  return x;
}

// Row pointer into the (virtually updated) KV cache. The freshly written slot
// is redirected to the fresh k/v row (inputs must not be mutated in place).
// pos is clamped to [0, live-1] so masked lanes never read out of bounds.
__device__ __forceinline__ const float* kv_row(const float* __restrict__ cache,
                                               const float* __restrict__ fresh,
                                               int base, int live, int slot,
                                               int hkv, int pos) {
  int pc  = (pos < live) ? pos : (live - 1);
  int idx = base + pc;
  const float* r = cache + ((size_t)idx * HKV_ + hkv) * DH_;
  if (slot >= 0 && idx == slot) r = fresh;
  return r;
}

__global__ __launch_bounds__(256) void fd_attn_kernel(
    const float* __restrict__ q,  const float* __restrict__ kn,
    const float* __restrict__ vn, const float* __restrict__ kc,
    const float* __restrict__ vc, const int* __restrict__ base_offsets,
    const int* __restrict__ live_counts, const int* __restrict__ slot_mapping,
    float* __restrict__ out) {

  __shared__ float lds_p[NW][16 * PS];   // per-wave P-tile transpose buffer
  __shared__ float lds_o[NW][G_][DH_];   // per-wave partial O
  __shared__ float lds_m[NW][G_];        // per-wave running max
  __shared__ float lds_s[NW][G_];        // per-wave running denom

  const int b    = blockIdx.x / HKV_;
  const int hkv  = blockIdx.x % HKV_;
  const int lane = threadIdx.x & 31;
  const int w    = threadIdx.x >> 5;
  const int nl   = lane & 15;            // low lane index (N position)
  const int hh   = lane >> 4;            // lane half (K-chunk position)

  const int base = base_offsets[b];
  const int live = live_counts[b];
  const int slot = slot_mapping[b];

  const float* kfresh = kn + ((size_t)b * HKV_ + hkv) * DH_;
  const float* vfresh = vn + ((size_t)b * HKV_ + hkv) * DH_;

  // ---- Preload Q as WMMA A fragments (heads 0..3 in rows 0..3, rest zero),
  // pre-scaled by 1/sqrt(DH). A layout: lane=(M row), element kk = j + 2*hh.
  v2f qa[32];
  {
    const int hq = (nl < G_) ? nl : 0;
    const float* qrow = q + ((size_t)b * H_ + hkv * G_ + hq) * DH_;
    const float scale = 0.08838834764831845f;  // 128^-0.5
    const v2f zz = {};
#pragma unroll
    for (int c = 0; c < 32; ++c) {
      v2f t = *(const v2f*)(qrow + 4 * c + 2 * hh);
      t = t * scale;
      qa[c] = (nl < G_) ? t : zz;
    }
  }

  // ---- Online-softmax state + O accumulator (C layout: head=r+8*hh, d=nl).
  v8f oacc[8];
#pragma unroll
  for (int dt = 0; dt < 8; ++dt) oacc[dt] = (v8f){};
  float mrun[8], srun[8];
#pragma unroll
  for (int r = 0; r < 8; ++r) { mrun[r] = -3.0e38f; srun[r] = 0.f; }

  const int ntiles = (live + 15) >> 4;

  for (int t = w; t < ntiles; t += NW) {
    const int kbase = t << 4;

    // This lane's K row (column key = nl), read in B layout straight from HBM.
    const float* krow = kv_row(kc, kfresh, base, live, slot, hkv, kbase + nl);
    // Prefetch this lane's K row of the next tile (speculative, gfx1250 path).
    __builtin_prefetch(kv_row(kc, kfresh, base, live, slot, hkv,
                              kbase + (NW << 4) + nl), 0, 1);

    // ---- S = Q * K^T : 32 chained fp32 WMMAs over DH=128.
    v8f sc = (v8f){};
#pragma unroll
    for (int c = 0; c < 32; ++c) {
      v2f kb = *(const v2f*)(krow + 4 * c + 2 * hh);
      sc = __builtin_amdgcn_wmma_f32_16x16x4_f32(
          false, qa[c], false, kb, (short)0, sc, false, false);
    }

    // ---- Mask + online softmax update (per head r, keys across 16 lanes).
    const bool vcol = (kbase + nl) < live;
    float p8[8], alpha[8];
#pragma unroll
    for (int r = 0; r < 8; ++r) {
      float x    = vcol ? sc[r] : -1.0e30f;
      float mt   = hmax16(x);
      float mnew = fmaxf(mrun[r], mt);
      alpha[r]   = __expf(mrun[r] - mnew);
      float pe   = __expf(x - mnew);
      srun[r]    = srun[r] * alpha[r] + hsum16(pe);
      mrun[r]    = mnew;
      p8[r]      = pe;
    }

    // ---- Transpose P through per-wave LDS (C layout -> A layout).
    float* pw = &lds_p[w][0];
#pragma unroll
    for (int r = 0; r < 8; ++r) pw[(r + 8 * hh) * PS + nl] = p8[r];

    v2f pa[4];
#pragma unroll
    for (int kt = 0; kt < 4; ++kt)
      pa[kt] = *(const v2f*)(pw + nl * PS + 4 * kt + 2 * hh);

    // ---- V row pointers for the B operand of P*V.
    const float* vr0[4];
    const float* vr1[4];
#pragma unroll
    for (int kt = 0; kt < 4; ++kt) {
      vr0[kt] = kv_row(vc, vfresh, base, live, slot, hkv, kbase + 4 * kt + 2 * hh);
      vr1[kt] = kv_row(vc, vfresh, base, live, slot, hkv, kbase + 4 * kt + 2 * hh + 1);
    }

    // ---- O = O*alpha + P*V : 32 WMMAs per tile (8 d-tiles x 4 k-steps).
#pragma unroll
    for (int dt = 0; dt < 8; ++dt) {
      v8f o = oacc[dt];
#pragma unroll
      for (int r = 0; r < 8; ++r) o[r] *= alpha[r];
#pragma unroll
      for (int kt = 0; kt < 4; ++kt) {
        v2f vb;
        vb.x = vr0[kt][16 * dt + nl];
        vb.y = vr1[kt][16 * dt + nl];
        o = __builtin_amdgcn_wmma_f32_16x16x4_f32(
            false, pa[kt], false, vb, (short)0, o, false, false);
      }
      oacc[dt] = o;
    }
  }

  // ---- Publish per-wave partials.
  if (hh == 0) {
    if (nl == 0) {
#pragma unroll
      for (int g = 0; g < G_; ++g) {
        lds_m[w][g] = mrun[g];
        lds_s[w][g] = srun[g];
      }
    }
#pragma unroll
    for (int dt = 0; dt < 8; ++dt)
#pragma unroll
      for (int g = 0; g < G_; ++g)
        lds_o[w][g][dt * 16 + nl] = oacc[dt][g];
  }
  __syncthreads();

  // ---- Cross-wave log-sum-exp combine + write out.
  for (int pidx = threadIdx.x; pidx < G_ * DH_; pidx += blockDim.x) {
    const int g = pidx >> 7;
    const int d = pidx & 127;
    float m = -3.0e38f;
#pragma unroll
    for (int ww = 0; ww < NW; ++ww) m = fmaxf(m, lds_m[ww][g]);
    float ss = 0.f, acc = 0.f;
#pragma unroll
    for (int ww = 0; ww < NW; ++ww) {
      float e = __expf(lds_m[ww][g] - m);
      ss  += lds_s[ww][g] * e;
      acc += lds_o[ww][g][d] * e;
    }
    out[((size_t)b * H_ + hkv * G_ + g) * DH_ + d] = acc / ss;
  }
}

extern "C" void kernel_launch(void* const* d_in, const int* in_sizes, int n_in,
                              void* d_out, int out_size, void* d_ws, size_t ws_size,
                              hipStream_t stream) {
  const float* q  = (const float*)d_in[0];
  const float* k  = (const float*)d_in[1];
  const float* v  = (const float*)d_in[2];
  const float* kc = (const float*)d_in[3];
  const float* vc = (const float*)d_in[4];
  const int* base_offsets = (const int*)d_in[5];
  const int* live_counts  = (const int*)d_in[6];
  const int* slot_mapping = (const int*)d_in[7];
  float* out = (float*)d_out;

  dim3 grid(B_ * HKV_);
  dim3 block(256);
  fd_attn_kernel<<<grid, block, 0, stream>>>(q, k, v, kc, vc, base_offsets,
                                             live_counts, slot_mapping, out);
}